// TopLineEncoder_36301063586581
// MI455X (gfx1250) — compile-verified
//
#include <hip/hip_runtime.h>

#define NLE 80
#define VOCAB 256
#define HID 128

typedef __attribute__((ext_vector_type(2))) float v2f;
typedef __attribute__((ext_vector_type(8))) float v8f;

__device__ __forceinline__ float elu1(float x) {
    return x > 0.0f ? x : (__expf(x) - 1.0f);
}

// Fused: out = ELU( ELU(gather_sum(W1, msg) + b1) @ W2 + b2 )
//
// One 256-thread block (8 wave32) owns 16 batch rows.
// Phase 1: each wave gathers 2 rows of the embedding sum (lane L owns float4
//          slice [4L,4L+4) of the 128-wide accumulator; every gathered W1 row
//          is a single coalesced 512B request, L2-resident since W1 = 10 MB
//          << 192 MB L2). Bias+ELU applied, result staged in an 8 KB LDS tile.
// Phase 2: each wave computes one 16x16 output tile via 32 chained
//          V_WMMA_F32_16X16X4_F32 (full fp32), A-fragments from LDS,
//          B-fragments from the L1-resident 64 KB W2.
__global__ __launch_bounds__(256) void fused_encoder_kernel(
    const long long* __restrict__ msg, const float* __restrict__ W1,
    const float* __restrict__ b1, const float* __restrict__ W2,
    const float* __restrict__ b2, float* __restrict__ out)
{
    __shared__ float hs[16 * HID];   // 8 KB h-tile

    const int lane = threadIdx.x & 31;
    const int wave = threadIdx.x >> 5;     // 0..7
    const int m0   = blockIdx.x * 16;

    // ---- Phase 1: embedding gather + sum + bias + ELU -> LDS ----
    const float4 bv = ((const float4*)b1)[lane];
    #pragma unroll
    for (int rr = 0; rr < 2; ++rr) {
        const int r   = wave * 2 + rr;     // row within tile, 0..15
        const int row = m0 + r;
        const long long* mrow = msg + (size_t)row * NLE;

        float4 acc = make_float4(0.f, 0.f, 0.f, 0.f);
        #pragma unroll 4
        for (int p = 0; p < NLE; ++p) {
            const int tok = (int)mrow[p];                    // wave-uniform
            const float4* wrow =
                (const float4*)(W1 + ((size_t)(p * VOCAB + tok)) * HID);
            const float4 v = wrow[lane];                     // 512B coalesced
            acc.x += v.x; acc.y += v.y; acc.z += v.z; acc.w += v.w;
        }
        acc.x = elu1(acc.x + bv.x);
        acc.y = elu1(acc.y + bv.y);
        acc.z = elu1(acc.z + bv.z);
        acc.w = elu1(acc.w + bv.w);
        ((float4*)(hs + r * HID))[lane] = acc;
    }
    __syncthreads();

    // ---- Phase 2: 16x16 fp32 WMMA GEMM + bias + ELU -> out ----
    const int n0    = wave * 16;           // column tile
    const int l15   = lane & 15;
    const int khalf = (lane >> 4) * 2;     // lanes 16-31 hold K+2, K+3

    const float* arow = hs + (size_t)l15 * HID;          // A: M = lane&15
    const float* bcol = W2 + (size_t)(n0 + l15);         // B: N = lane&15

    v8f c = {};
    #pragma unroll
    for (int k0 = 0; k0 < HID; k0 += 4) {
        v2f a, b;
        a.x = arow[k0 + khalf];                          // ds_load (b64 pair)
        a.y = arow[k0 + khalf + 1];
        b.x = bcol[(size_t)(k0 + khalf) * HID];
        b.y = bcol[(size_t)(k0 + khalf + 1) * HID];
        // 8 args: (neg_a, A, neg_b, B, c_mod, C, reuse_a, reuse_b)
        c = __builtin_amdgcn_wmma_f32_16x16x4_f32(
                false, a, false, b, (short)0, c, false, false);
    }

    // C/D layout: VGPR r holds M = r + 8*(lane>=16), N = lane&15.
    const float bias  = b2[n0 + l15];
    const int   mbase = m0 + (lane >> 4) * 8;
    #pragma unroll
    for (int r = 0; r < 8; ++r) {
        const float v = elu1(c[r] + bias);
        out[(size_t)(mbase + r) * HID + (n0 + l15)] = v;
    }
}

extern "C" void kernel_launch(void* const* d_in, const int* in_sizes, int n_in,
                              void* d_out, int out_size, void* d_ws, size_t ws_size,
                              hipStream_t stream) {
    const long long* msg = (const long long*)d_in[0];   // int64 [B,80]
    const float*     W1  = (const float*)d_in[1];       // [20480,128]
    const float*     b1  = (const float*)d_in[2];       // [128]
    const float*     W2  = (const float*)d_in[3];       // [128,128]
    const float*     b2  = (const float*)d_in[4];       // [128]
    float* out = (float*)d_out;                         // [B,128] fp32

    const int B = in_sizes[0] / NLE;                    // 16384

    fused_encoder_kernel<<<B / 16, 256, 0, stream>>>(msg, W1, b1, W2, b2, out);
}